// SparseAttention_68728066671275
// MI455X (gfx1250) — compile-verified
//
#include <hip/hip_runtime.h>

typedef __attribute__((ext_vector_type(2))) float v2f;
typedef __attribute__((ext_vector_type(4))) float v4f;
typedef __attribute__((ext_vector_type(8))) float v8f;

// Geometry from the reference: HEADS=8, EMB=64, x is (8, 8192, 1) -> scalar per token.
// out[token][h][e] = x_t * softmax_g( x_t^2 * scale * G[h][g] ) @ Wv_r[g][e]
// where G = Wq_r @ Wk_r^T is a constant 8x8, Wv_r is the constant 8x64 value matrix.
// Computed transposed (out^T = Wv^T @ S^T) so each lane's 8 D-VGPRs are 8
// contiguous floats -> two global_store_b128 per 16-wide e-tile.

__global__ __launch_bounds__(256) void sparse_attn_wmma(
    const float* __restrict__ x,
    const float* __restrict__ Wq,
    const float* __restrict__ Wk,
    const float* __restrict__ Wv,
    float* __restrict__ out,
    int pairs)
{
    __shared__ float Gs[64];

    const int tid = threadIdx.x;

    // ---- Per-workgroup preamble: G[h][g] = dot(Wq_row_h, Wk_row_g), 8x8 ----
    if (tid < 64) {
        const int h = tid >> 3, g = tid & 7;
        const float* qr = Wq + h * 64;
        const float* kr = Wk + g * 64;
        float acc = 0.f;
        #pragma unroll
        for (int e = 0; e < 64; ++e) acc += qr[e] * kr[e];
        Gs[tid] = acc;
    }
    __syncthreads();

    const int lane = tid & 31;
    const int wave = tid >> 5;
    const int lmod = lane & 15;      // tile row/col index 0..15
    const int lhi  = lane >> 4;      // half-wave selector (K group / M group)
    const bool hi  = (lhi != 0);     // loop-invariant select condition
    const int h    = lmod & 7;       // head of this lane's softmax row
    const int tl   = lmod >> 3;      // token-in-pair of this lane's softmax row

    // G row for this lane's head; pre-subtract the row max (scores are
    // s*G with s = x^2*scale >= 0, so row max is s*Gmax).
    float Gd[8];
    {
        float Grow[8];
        #pragma unroll
        for (int g = 0; g < 8; ++g) Grow[g] = Gs[h * 8 + g];
        float Gmax = Grow[0];
        #pragma unroll
        for (int g = 1; g < 8; ++g) Gmax = fmaxf(Gmax, Grow[g]);
        #pragma unroll
        for (int g = 0; g < 8; ++g) Gd[g] = Grow[g] - Gmax;
    }

    // ---- Constant A operand: Wv^T tiles. A layout (16x4 f32):
    //   lane L: M = L%16, VGPR i holds K = i + 2*(L/16)
    // A[e_tile t][kgroup kk]: A[m][k] = Wv[g = k + 4*kk][e = 16*t + m]
    v2f Av[4][2];
    #pragma unroll
    for (int t = 0; t < 4; ++t) {
        #pragma unroll
        for (int kk = 0; kk < 2; ++kk) {
            const int g0 = 2 * lhi + 4 * kk;
            v2f a;
            a.x = Wv[(g0 + 0) * 64 + 16 * t + lmod];
            a.y = Wv[(g0 + 1) * 64 + 16 * t + lmod];
            Av[t][kk] = a;
        }
    }

    const float scale = 0.35355339059327373f; // 1/sqrt(8)
    const int wavesPerBlock = blockDim.x >> 5;
    const int totalWaves = gridDim.x * wavesPerBlock;
    const int gw = blockIdx.x * wavesPerBlock + wave;

    for (int pair = gw; pair < pairs; pair += totalWaves) {
        // x scalar for this lane's row-token (lanes with same lmod share it)
        const float xt = x[pair * 2 + tl];
        const float s  = xt * xt * scale;

        // ---- softmax numerators (max already folded out) ----
        float e0 = __expf(s * Gd[0]);
        float e1 = __expf(s * Gd[1]);
        float e2 = __expf(s * Gd[2]);
        float e3 = __expf(s * Gd[3]);
        float e4 = __expf(s * Gd[4]);
        float e5 = __expf(s * Gd[5]);
        float e6 = __expf(s * Gd[6]);
        float e7 = __expf(s * Gd[7]);
        const float sum = ((e0 + e1) + (e2 + e3)) + ((e4 + e5) + (e6 + e7));
        // fold the trailing x_t multiply into the probabilities
        const float inv = xt * __builtin_amdgcn_rcpf(sum);

        // B operand = S^T (8x16): lane L holds column n = L%16,
        // VGPR i holds K = i + 2*(L/16); two K-groups cover g=0..7.
        // lanes 0-15 use {e0,e1,e4,e5}; lanes 16-31 use {e2,e3,e6,e7}.
        v2f B1, B2;
        B1.x = (hi ? e2 : e0) * inv;
        B1.y = (hi ? e3 : e1) * inv;
        B2.x = (hi ? e6 : e4) * inv;
        B2.y = (hi ? e7 : e5) * inv;

        // Output base for this lane: token tl, head h; e offset 8*lhi within tile
        float* dst = out + (size_t)pair * 1024 + (size_t)tl * 512 + h * 64 + 8 * lhi;

        #pragma unroll
        for (int t = 0; t < 4; ++t) {
            v8f c = {};
            // D = A(Wv^T) x B(S^T): two K=4 steps accumulate g=0..7
            c = __builtin_amdgcn_wmma_f32_16x16x4_f32(
                    false, Av[t][0], false, B1, (short)0, c, false, false);
            c = __builtin_amdgcn_wmma_f32_16x16x4_f32(
                    false, Av[t][1], false, B2, (short)0, c, false, false);
            // Lane's D VGPRs r=0..7 are e = 16*t + 8*lhi + r -> contiguous 8 floats
            v4f lo, hi4;
            lo.x  = c[0]; lo.y  = c[1]; lo.z  = c[2]; lo.w  = c[3];
            hi4.x = c[4]; hi4.y = c[5]; hi4.z = c[6]; hi4.w = c[7];
            *(v4f*)(dst + 16 * t)     = lo;
            *(v4f*)(dst + 16 * t + 4) = hi4;
        }
    }
}

extern "C" void kernel_launch(void* const* d_in, const int* in_sizes, int n_in,
                              void* d_out, int out_size, void* d_ws, size_t ws_size,
                              hipStream_t stream) {
    const float* x  = (const float*)d_in[0];
    const float* Wq = (const float*)d_in[1];
    const float* Wk = (const float*)d_in[2];
    const float* Wv = (const float*)d_in[3];
    float* out = (float*)d_out;

    const int tokens = in_sizes[0]; // b*n*hin = 65536
    const int pairs  = tokens / 2;

    dim3 block(256);   // 8 waves per workgroup (wave32)
    dim3 grid(512);    // 4096 waves -> 8 token-pairs per wave (amortizes preamble)
    sparse_attn_wmma<<<grid, block, 0, stream>>>(x, Wq, Wk, Wv, out, pairs);
}